// SABlock_CrossMamba_70944269795624
// MI455X (gfx1250) — compile-verified
//
#include <hip/hip_runtime.h>
#include <hip/hip_bf16.h>
#include <math.h>

// ---------------------------------------------------------------------------
// SABlock_CrossMamba for MI455X (gfx1250, wave32).
// fp32 end-to-end; GEMMs use V_WMMA_F32_16X16X4_F32; big GEMMs stage their
// B panel in LDS via GLOBAL_LOAD_ASYNC_TO_LDS_B128 (ASYNCcnt path).
// Dims (fixed by reference): B=4, L=512, D=128, Di=256, S=64, R=8, K=4, H=256.
// ---------------------------------------------------------------------------

typedef __attribute__((ext_vector_type(2))) float v2f;
typedef __attribute__((ext_vector_type(8))) float v8f;

#define T_TOK   2048
#define DMODEL  128
#define DINNER  256
#define SSTATE  64
#define RRANK   8
#define HFFN    256
#define XPPAD   144      // R + 2S = 136 padded to multiple of 16
#define NPROJ   512      // 2*Di
#define KB      32       // K-block staged in LDS per iteration

// ---------------------------------------------------------------------------
// Big-tile fp32 WMMA GEMM: C[MxN] = A[MxK] * B  (B KxN, or NxK if TRANSB).
// Block = 128 threads = 4 waves; each wave owns a 16(M)x64(N) strip, so the
// block covers 64x64 and the shared KBx64 B panel is staged once in LDS by
// async global->LDS loads, then read by all 4 waves. Requires N%64==0, K%32==0.
// ---------------------------------------------------------------------------
template <bool TRANSB, bool ACCUM>
__global__ __launch_bounds__(128)
void k_gemm_f32_t64(const float* __restrict__ A, const float* __restrict__ B,
                    float* __restrict__ C, int M, int N, int Kd) {
  __shared__ float bpan[KB * 64];     // !TRANSB: [k][n]  |  TRANSB: [n][k]
  const int tidx  = threadIdx.x;
  const int lane  = tidx & 31;
  const int wave  = tidx >> 5;
  const int tileN = blockIdx.x << 6;                 // 64-wide N strip
  const int tileM = ((blockIdx.y << 2) + wave) << 4; // 16-wide M strip per wave
  const int half  = lane >> 4;                       // 0: K{0,1}, 1: K{2,3}
  const int l     = lane & 15;
  const int kst   = half << 1;
  const int rowA  = tileM + l;

  v8f acc[4];
  const v8f vzero = {0.f, 0.f, 0.f, 0.f, 0.f, 0.f, 0.f, 0.f};
#pragma unroll
  for (int j = 0; j < 4; ++j) {
    acc[j] = vzero;
    if (ACCUM) {
#pragma unroll
      for (int r = 0; r < 8; ++r)
        acc[j][r] = C[(size_t)(tileM + half * 8 + r) * N + tileN + 16 * j + l];
    }
  }

  // per-lane pieces of the cooperative B-panel stage (16B per async load)
  const int nseg = (tidx & 15) << 2;   // !TRANSB: 4 consecutive n
  const int kk0  = tidx >> 4;          // !TRANSB: k row 0..7 (+8 per round)
  const int kseg = (tidx & 7) << 2;    // TRANSB: 4 consecutive k
  const int n0   = tidx >> 3;          // TRANSB: n 0..15 (+16 per round)

  for (int kp = 0; kp < Kd; kp += KB) {
    if (!TRANSB) {
      const float* sbase = B + (size_t)kp * N + tileN;   // uniform
#pragma unroll
      for (int r = 0; r < 4; ++r) {
        const int kk = kk0 + (r << 3);
        unsigned lds  = (unsigned)(uintptr_t)&bpan[kk * 64 + nseg];
        unsigned voff = (unsigned)((kk * N + nseg) * 4); // bytes, 16B aligned
        asm volatile("global_load_async_to_lds_b128 %0, %1, %2"
                     :: "v"(lds), "v"(voff), "s"(sbase) : "memory");
      }
    } else {
      const float* sbase = B + (size_t)tileN * Kd + kp;  // uniform
#pragma unroll
      for (int r = 0; r < 4; ++r) {
        const int n = n0 + (r << 4);
        unsigned lds  = (unsigned)(uintptr_t)&bpan[n * KB + kseg];
        unsigned voff = (unsigned)((n * Kd + kseg) * 4);
        asm volatile("global_load_async_to_lds_b128 %0, %1, %2"
                     :: "v"(lds), "v"(voff), "s"(sbase) : "memory");
      }
    }
    asm volatile("s_wait_asynccnt 0x0" ::: "memory");
    __syncthreads();

#pragma unroll
    for (int ks = 0; ks < KB; ks += 4) {
      v2f a;
      const float* ap = A + (size_t)rowA * Kd + kp + ks + kst;
      a.x = ap[0];
      a.y = ap[1];
#pragma unroll
      for (int j = 0; j < 4; ++j) {
        v2f b;
        if (!TRANSB) {
          b.x = bpan[(ks + kst) * 64 + 16 * j + l];
          b.y = bpan[(ks + kst + 1) * 64 + 16 * j + l];
        } else {
          b.x = bpan[(16 * j + l) * KB + ks + kst];
          b.y = bpan[(16 * j + l) * KB + ks + kst + 1];
        }
        acc[j] = __builtin_amdgcn_wmma_f32_16x16x4_f32(false, a, false, b,
                                                       (short)0, acc[j],
                                                       false, false);
      }
    }
    __syncthreads();
  }

#pragma unroll
  for (int j = 0; j < 4; ++j)
#pragma unroll
    for (int r = 0; r < 8; ++r)
      C[(size_t)(tileM + half * 8 + r) * N + tileN + 16 * j + l] = acc[j][r];
}

// ---------------------------------------------------------------------------
// Small fp32 WMMA GEMM (one 16x16 tile per wave); used for N=144 (x_proj).
// ---------------------------------------------------------------------------
template <bool TRANSB, bool ACCUM>
__global__ __launch_bounds__(128)
void k_gemm_f32_t16(const float* __restrict__ A, const float* __restrict__ B,
                    float* __restrict__ C, int M, int N, int Kd) {
  const int lane  = threadIdx.x & 31;
  const int wave  = threadIdx.x >> 5;
  const int tileN = blockIdx.x << 4;
  const int tileM = ((blockIdx.y << 2) + wave) << 4;
  if (tileM >= M) return;                 // uniform per wave: EXEC stays full
  const int half = lane >> 4;
  const int l    = lane & 15;
  const int kst  = half << 1;
  const int rowA = tileM + l;

  v8f acc = {0.f, 0.f, 0.f, 0.f, 0.f, 0.f, 0.f, 0.f};
  if (ACCUM) {
#pragma unroll
    for (int r = 0; r < 8; ++r)
      acc[r] = C[(size_t)(tileM + half * 8 + r) * N + tileN + l];
  }
  for (int k0 = 0; k0 < Kd; k0 += 4) {
    v2f a, b;
    const float* ap = A + (size_t)rowA * Kd + k0 + kst;
    a.x = ap[0];
    a.y = ap[1];
    __builtin_prefetch(ap + 64, 0, 1);    // global_prefetch_b8
    if (TRANSB) {
      const float* bp = B + (size_t)(tileN + l) * Kd + k0 + kst;
      b.x = bp[0];
      b.y = bp[1];
    } else {
      b.x = B[(size_t)(k0 + kst) * N + tileN + l];
      b.y = B[(size_t)(k0 + kst + 1) * N + tileN + l];
    }
    acc = __builtin_amdgcn_wmma_f32_16x16x4_f32(false, a, false, b,
                                                (short)0, acc, false, false);
  }
#pragma unroll
  for (int r = 0; r < 8; ++r)
    C[(size_t)(tileM + half * 8 + r) * N + tileN + l] = acc[r];
}

// ---------------------------------------------------------------------------
// LayerNorm over D=128 per token; optional residual addend: y = addend + LN(x)
// ---------------------------------------------------------------------------
__global__ __launch_bounds__(128)
void k_layernorm(const float* __restrict__ xin, const float* __restrict__ wt,
                 const float* __restrict__ bs, const float* __restrict__ addend,
                 float* __restrict__ yout) {
  const int t = blockIdx.x;
  const int i = threadIdx.x;          // 128
  __shared__ float r1[128], r2[128];
  float v = xin[(size_t)t * DMODEL + i];
  r1[i] = v;
  r2[i] = v * v;
  __syncthreads();
  for (int st = 64; st > 0; st >>= 1) {
    if (i < st) { r1[i] += r1[i + st]; r2[i] += r2[i + st]; }
    __syncthreads();
  }
  const float mu  = r1[0] * (1.f / DMODEL);
  const float var = r2[0] * (1.f / DMODEL) - mu * mu;
  float o = (v - mu) * rsqrtf(var + 1e-5f) * wt[i] + bs[i];
  if (addend) o += addend[(size_t)t * DMODEL + i];
  yout[(size_t)t * DMODEL + i] = o;
}

// ---------------------------------------------------------------------------
// Causal depthwise conv (K=4, resets at chunk boundaries) + bias + SiLU.
// ---------------------------------------------------------------------------
__global__ void k_conv_silu(const float* __restrict__ xz,
                            const float* __restrict__ cw,
                            const float* __restrict__ cb,
                            float* __restrict__ xh, int Lc) {
  int idx = blockIdx.x * blockDim.x + threadIdx.x;
  if (idx >= T_TOK * DINNER) return;
  const int d = idx & (DINNER - 1);
  const int t = idx >> 8;
  const int pos = t % Lc;
  float acc = cb[d];
#pragma unroll
  for (int j = 0; j < 4; ++j) {
    if (pos - 3 + j >= 0)
      acc += cw[d * 4 + j] * xz[(size_t)(t - 3 + j) * NPROJ + d];
  }
  xh[idx] = acc / (1.f + __expf(-acc));   // silu
}

// dt = softplus(dbl[:, :8] @ dt_proj_w + dt_proj_b)
__global__ void k_dt(const float* __restrict__ dbl, const float* __restrict__ wt,
                     const float* __restrict__ bt, float* __restrict__ dt) {
  int idx = blockIdx.x * blockDim.x + threadIdx.x;
  if (idx >= T_TOK * DINNER) return;
  const int d = idx & (DINNER - 1);
  const int t = idx >> 8;
  float a = bt[d];
#pragma unroll
  for (int r = 0; r < RRANK; ++r)
    a += dbl[(size_t)t * XPPAD + r] * wt[r * DINNER + d];
  dt[idx] = (a > 20.f) ? a : log1pf(__expf(a));
}

// Copy Cm (columns 72..135 of dbl) into the output tensor for this pass.
__global__ void k_copyC(const float* __restrict__ dbl, float* __restrict__ Cout) {
  int idx = blockIdx.x * blockDim.x + threadIdx.x;
  if (idx >= T_TOK * SSTATE) return;
  const int s = idx & (SSTATE - 1);
  const int t = idx >> 6;
  Cout[idx] = dbl[(size_t)t * XPPAD + RRANK + SSTATE + s];
}

// ---------------------------------------------------------------------------
// Selective scan: one block (256 threads) per sequence, thread d keeps h[64]
// and A_row[64] in registers; B/C rows broadcast through LDS each step.
// ---------------------------------------------------------------------------
__global__ __launch_bounds__(256)
void k_scan(const float* __restrict__ dbl, const float* __restrict__ dt,
            const float* __restrict__ xh, const float* __restrict__ xz,
            const float* __restrict__ A_log, const float* __restrict__ Dp,
            float* __restrict__ y, int Lc) {
  const int seq = blockIdx.x;
  const int d   = threadIdx.x;
  const int t0  = seq * Lc;
  __shared__ float Bs[SSTATE], Cs[SSTATE];
  float h[SSTATE], Ar[SSTATE];
#pragma unroll
  for (int s = 0; s < SSTATE; ++s) {
    h[s]  = 0.f;
    Ar[s] = -__expf(A_log[d * SSTATE + s]);
  }
  const float dpar = Dp[d];
  for (int tt = 0; tt < Lc; ++tt) {
    const int t = t0 + tt;
    if (d < 64)       Bs[d]      = dbl[(size_t)t * XPPAD + RRANK + d];
    else if (d < 128) Cs[d - 64] = dbl[(size_t)t * XPPAD + RRANK + SSTATE + (d - 64)];
    __syncthreads();
    const float dtv = dt[(size_t)t * DINNER + d];
    const float xv  = xh[(size_t)t * DINNER + d];
    float acc = 0.f;
#pragma unroll
    for (int s = 0; s < SSTATE; ++s) {
      h[s] = __expf(dtv * Ar[s]) * h[s] + dtv * Bs[s] * xv;
      acc += h[s] * Cs[s];
    }
    const float z  = xz[(size_t)t * NPROJ + DINNER + d];
    const float sz = z / (1.f + __expf(-z));
    y[(size_t)t * DINNER + d] = (acc + dpar * xv) * sz;
    __syncthreads();
  }
}

// ---------------------------------------------------------------------------
// BatchNorm over (b,n): stats then apply (optional relu / residual add).
// ---------------------------------------------------------------------------
__global__ __launch_bounds__(256)
void k_bn_stats(const float* __restrict__ h, float* __restrict__ mu,
                float* __restrict__ rstd, int T, int C) {
  const int c   = blockIdx.x;
  const int tid = threadIdx.x;
  __shared__ float r1[256], r2[256];
  float s = 0.f, s2 = 0.f;
  for (int t = tid; t < T; t += 256) {
    float v = h[(size_t)t * C + c];
    s += v; s2 += v * v;
  }
  r1[tid] = s; r2[tid] = s2;
  __syncthreads();
  for (int st = 128; st > 0; st >>= 1) {
    if (tid < st) { r1[tid] += r1[tid + st]; r2[tid] += r2[tid + st]; }
    __syncthreads();
  }
  if (tid == 0) {
    float m = r1[0] / (float)T;
    mu[c]   = m;
    rstd[c] = rsqrtf(r2[0] / (float)T - m * m + 1e-5f);
  }
}

__global__ void k_bn_apply(const float* __restrict__ h, const float* __restrict__ mu,
                           const float* __restrict__ rstd, const float* __restrict__ g,
                           const float* __restrict__ b, const float* __restrict__ addend,
                           float* __restrict__ out, int total, int C, int relu) {
  int idx = blockIdx.x * blockDim.x + threadIdx.x;
  if (idx >= total) return;
  const int c = idx % C;
  float v = (h[idx] - mu[c]) * rstd[c] * g[c] + b[c];
  if (relu) v = fmaxf(v, 0.f);
  if (addend) v += addend[idx];
  out[idx] = v;
}

// (B,L,H) <-> (B,H,L) transposes for contiguous FFT rows.
__global__ void k_tr_fwd(const float* __restrict__ in, float* __restrict__ out) {
  int idx = blockIdx.x * blockDim.x + threadIdx.x;
  if (idx >= 4 * 512 * 256) return;
  const int h = idx & 255;
  const int n = (idx >> 8) & 511;
  const int b = idx >> 17;
  out[(((size_t)b << 8) + h) * 512 + n] = in[idx];
}
__global__ void k_tr_bwd(const float* __restrict__ in, float* __restrict__ out) {
  int idx = blockIdx.x * blockDim.x + threadIdx.x;
  if (idx >= 4 * 256 * 512) return;
  const int n = idx & 511;
  const int h = (idx >> 9) & 255;
  const int b = idx >> 17;
  out[(((size_t)b << 9) + n) * 256 + h] = in[idx];
}

// ---------------------------------------------------------------------------
// 512-point radix-2 DIT FFT in LDS (input bit-reversed -> natural order out).
// ---------------------------------------------------------------------------
__device__ __forceinline__ void fft512_lds(float* re, float* im, int tid) {
  for (int len = 2; len <= 512; len <<= 1) {
    const int half = len >> 1;
    const int grp  = tid / half;
    const int pos  = tid - grp * half;
    const int i0   = grp * len + pos;
    const int i1   = i0 + half;
    float sn, cs;
    __sincosf(-6.283185307179586f * (float)pos / (float)len, &sn, &cs);
    const float tr = re[i1] * cs - im[i1] * sn;
    const float ti = re[i1] * sn + im[i1] * cs;
    re[i1] = re[i0] - tr; im[i1] = im[i0] - ti;
    re[i0] += tr;         im[i0] += ti;
    __syncthreads();
  }
}

__global__ __launch_bounds__(256)
void k_freq(float* __restrict__ ht, const float* __restrict__ rm,
            const float* __restrict__ imat, const float* __restrict__ rb,
            const float* __restrict__ ib) {
  __shared__ float re[512], im[512];
  const int bh  = blockIdx.x;
  const int hh  = bh & 255;
  float* row    = ht + (size_t)bh * 512;
  const int tid = threadIdx.x;
  for (int i = tid; i < 512; i += 256) {
    const int r = (int)(__brev((unsigned)i) >> 23);
    re[i] = row[r];
    im[i] = 0.f;
  }
  __syncthreads();
  fft512_lds(re, im, tid);
  const float sc  = 0.04419417382415922f;  // 1/sqrt(512)
  const float rd  = rm[hh * 256 + hh];
  const float ig  = imat[hh * 256 + hh];
  const float rbv = rb[hh], ibv = ib[hh];
  for (int i = tid; i < 512; i += 256) {
    const float xr = re[i] * sc, xi = im[i] * sc;
    const float nr = fmaxf(xr * rd - xi * ig + rbv, 0.f);
    const float ni = fmaxf(xi * rd + xr * ig + ibv, 0.f);
    re[i] = nr;
    im[i] = -ni;                            // conjugate for inverse transform
  }
  __syncthreads();
  for (int i = tid; i < 512; i += 256) {    // in-place bit-reverse permute
    const int r = (int)(__brev((unsigned)i) >> 23);
    if (i < r) {
      float a = re[i]; re[i] = re[r]; re[r] = a;
      float b = im[i]; im[i] = im[r]; im[r] = b;
    }
  }
  __syncthreads();
  fft512_lds(re, im, tid);
  for (int i = tid; i < 512; i += 256)
    row[i] = re[i] * sc;                    // Re[ifft_ortho]
}

__global__ void k_zero(float* __restrict__ p, int n) {
  int i = blockIdx.x * blockDim.x + threadIdx.x;
  if (i < n) p[i] = 0.f;
}

__global__ void k_pad_w(const float* __restrict__ w, float* __restrict__ wp) {
  int idx = blockIdx.x * blockDim.x + threadIdx.x;
  if (idx >= DINNER * XPPAD) return;
  const int j = idx % XPPAD;
  const int c = idx / XPPAD;
  wp[idx] = (j < RRANK + 2 * SSTATE) ? w[c * (RRANK + 2 * SSTATE) + j] : 0.f;
}

// ---------------------------------------------------------------------------
extern "C" void kernel_launch(void* const* d_in, const int* in_sizes, int n_in,
                              void* d_out, int out_size, void* d_ws, size_t ws_size,
                              hipStream_t stream) {
  (void)in_sizes; (void)n_in; (void)out_size; (void)ws_size;
  const float* x         = (const float*)d_in[0];
  const float* ln_w      = (const float*)d_in[1];
  const float* ln_b      = (const float*)d_in[2];
  const float* in_proj_w = (const float*)d_in[3];
  const float* conv_w    = (const float*)d_in[4];
  const float* conv_b    = (const float*)d_in[5];
  const float* x_proj_w  = (const float*)d_in[6];
  const float* dt_proj_w = (const float*)d_in[7];
  const float* dt_proj_b = (const float*)d_in[8];
  const float* A_log     = (const float*)d_in[9];
  const float* D_param   = (const float*)d_in[10];
  const float* out_proj_w= (const float*)d_in[11];
  const float* norm1_w   = (const float*)d_in[12];
  const float* norm1_b   = (const float*)d_in[13];
  const float* norm2_w   = (const float*)d_in[14];
  const float* norm2_b   = (const float*)d_in[15];
  const float* fc1_w     = (const float*)d_in[16];
  const float* bn1_g     = (const float*)d_in[17];
  const float* bn1_b     = (const float*)d_in[18];
  const float* r_mat     = (const float*)d_in[19];
  const float* i_mat     = (const float*)d_in[20];
  const float* rb        = (const float*)d_in[21];
  const float* ib        = (const float*)d_in[22];
  const float* fc2_w     = (const float*)d_in[23];
  const float* bn2_g     = (const float*)d_in[24];
  const float* bn2_b     = (const float*)d_in[25];
  float* out = (float*)d_out;

  float* w = (float*)d_ws;
  size_t off = 0;
  auto alloc = [&](size_t n) { float* p = w + off; off += n; return p; };
  float* xz     = alloc((size_t)T_TOK * NPROJ);   // 2048x512
  float* xn     = alloc((size_t)T_TOK * DMODEL);
  float* xh     = alloc((size_t)T_TOK * DINNER);
  float* dbl    = alloc((size_t)T_TOK * XPPAD);
  float* dtv    = alloc((size_t)T_TOK * DINNER);
  float* ybuf   = alloc((size_t)T_TOK * DINNER);
  float* xo_acc = alloc((size_t)T_TOK * DMODEL);
  float* xo     = alloc((size_t)T_TOK * DMODEL);
  float* xn2    = alloc((size_t)T_TOK * DMODEL);
  float* h1     = alloc((size_t)T_TOK * HFFN);
  float* ht     = alloc((size_t)T_TOK * HFFN);
  float* ffn    = alloc((size_t)T_TOK * DMODEL);
  float* xpw    = alloc((size_t)DINNER * XPPAD);
  float* mu     = alloc(256);
  float* rstd   = alloc(256);

  const dim3 b128(128), b256(256);

  // Shared across all 4 passes: LN(x) and in_proj.
  k_layernorm<<<T_TOK, b128, 0, stream>>>(x, ln_w, ln_b, nullptr, xn);
  k_gemm_f32_t64<false, false><<<dim3(NPROJ / 64, T_TOK / 64), b128, 0, stream>>>(
      xn, in_proj_w, xz, T_TOK, NPROJ, DMODEL);
  k_pad_w<<<(DINNER * XPPAD + 255) / 256, b256, 0, stream>>>(x_proj_w, xpw);
  k_zero<<<(T_TOK * DMODEL + 255) / 256, b256, 0, stream>>>(xo_acc, T_TOK * DMODEL);

  const int Lcs[4] = {512, 256, 128, 64};
  for (int p = 0; p < 4; ++p) {
    const int Lc = Lcs[p];
    k_conv_silu<<<(T_TOK * DINNER + 255) / 256, b256, 0, stream>>>(xz, conv_w, conv_b, xh, Lc);
    k_gemm_f32_t16<false, false><<<dim3(XPPAD / 16, T_TOK / 64), b128, 0, stream>>>(
        xh, xpw, dbl, T_TOK, XPPAD, DINNER);
    k_dt<<<(T_TOK * DINNER + 255) / 256, b256, 0, stream>>>(dbl, dt_proj_w, dt_proj_b, dtv);
    k_copyC<<<(T_TOK * SSTATE + 255) / 256, b256, 0, stream>>>(
        dbl, out + (size_t)T_TOK * DMODEL + (size_t)p * T_TOK * SSTATE);
    k_scan<<<T_TOK / Lc, b256, 0, stream>>>(dbl, dtv, xh, xz, A_log, D_param, ybuf, Lc);
    k_gemm_f32_t64<false, true><<<dim3(DMODEL / 64, T_TOK / 64), b128, 0, stream>>>(
        ybuf, out_proj_w, xo_acc, T_TOK, DMODEL, DINNER);
  }

  // xo = x + LN(x_o); xn2 = LN(xo)
  k_layernorm<<<T_TOK, b128, 0, stream>>>(xo_acc, norm1_w, norm1_b, x, xo);
  k_layernorm<<<T_TOK, b128, 0, stream>>>(xo, norm2_w, norm2_b, nullptr, xn2);

  // freq_ffn: fc1 (B transposed) -> bn1+relu -> FFT path -> fc2 (B transposed) -> bn2 + residual
  k_gemm_f32_t64<true, false><<<dim3(HFFN / 64, T_TOK / 64), b128, 0, stream>>>(
      xn2, fc1_w, h1, T_TOK, HFFN, DMODEL);
  k_bn_stats<<<HFFN, b256, 0, stream>>>(h1, mu, rstd, T_TOK, HFFN);
  k_bn_apply<<<(T_TOK * HFFN + 255) / 256, b256, 0, stream>>>(
      h1, mu, rstd, bn1_g, bn1_b, nullptr, h1, T_TOK * HFFN, HFFN, 1);
  k_tr_fwd<<<(T_TOK * HFFN + 255) / 256, b256, 0, stream>>>(h1, ht);
  k_freq<<<4 * HFFN, b256, 0, stream>>>(ht, r_mat, i_mat, rb, ib);
  k_tr_bwd<<<(T_TOK * HFFN + 255) / 256, b256, 0, stream>>>(ht, h1);
  k_gemm_f32_t64<true, false><<<dim3(DMODEL / 64, T_TOK / 64), b128, 0, stream>>>(
      h1, fc2_w, ffn, T_TOK, DMODEL, HFFN);
  k_bn_stats<<<DMODEL, b256, 0, stream>>>(ffn, mu, rstd, T_TOK, DMODEL);
  k_bn_apply<<<(T_TOK * DMODEL + 255) / 256, b256, 0, stream>>>(
      ffn, mu, rstd, bn2_g, bn2_b, xo, out, T_TOK * DMODEL, DMODEL, 0);
}